// BoundaryLoss_28089086116184
// MI455X (gfx1250) — compile-verified
//
#include <hip/hip_runtime.h>
#include <hip/hip_bf16.h>
#include <math.h>

// Problem constants: pred [4,1,512,512] f32, target [4,1,512,512] int, out = scalar f32.
#define B_ 4
#define H_ 512
#define W_ 512
#define ROWS_ (B_ * H_)          // 2048
#define NELEM_ (B_ * H_ * W_)    // 1048576

typedef float v2f __attribute__((ext_vector_type(2)));
typedef float v8f __attribute__((ext_vector_type(8)));

// ---------------------------------------------------------------------------
// K0: zero the per-batch has-foreground flags (d_ws is poisoned once, never
// re-poisoned between replays, so we must re-init every call).
// ---------------------------------------------------------------------------
__global__ void bl_init_flags(int* __restrict__ hasfg) {
    if (threadIdx.x < B_) hasfg[threadIdx.x] = 0;
}

// ---------------------------------------------------------------------------
// K1: Phase 1 — per-column 1D distance along the row axis.
// One thread per (b,w) column (2048 threads). Down pass writes h-last into
// g2, up pass reads it back (same thread, same addresses: ordered), clamps to
// big=H+W=1024 and squares. Matches the reference cummax/cummin construction.
// ---------------------------------------------------------------------------
__global__ void bl_edt_cols(const int* __restrict__ target,
                            float* __restrict__ g2,
                            int* __restrict__ hasfg) {
    const int tid = blockIdx.x * blockDim.x + threadIdx.x;   // 0..2047
    if (tid >= B_ * W_) return;
    const int b = tid >> 9;          // tid / W_
    const int w = tid & (W_ - 1);    // tid % W_
    const int base = b * H_ * W_ + w;
    const float BIG = (float)(H_ + W_);   // 1024, > max possible distance

    float last = -BIG;
    int any = 0;
    for (int h = 0; h < H_; ++h) {
        const int fg = target[base + h * W_] > 0;
        any |= fg;
        if (fg) last = (float)h;
        g2[base + h * W_] = (float)h - last;        // distance to fg above
    }
    float nxt = 3.0f * BIG;
    for (int h = H_ - 1; h >= 0; --h) {
        const int fg = target[base + h * W_] > 0;
        if (fg) nxt = (float)h;
        const float gd = g2[base + h * W_];
        float g = fminf(fminf(gd, nxt - (float)h), BIG);
        g2[base + h * W_] = g * g;
    }
    if (any) atomicOr(&hasfg[b], 1);   // bitwise -> order-independent, deterministic
}

// ---------------------------------------------------------------------------
// K2: Phase 2 + loss partials. One block per (b,h) row. g2 row staged in LDS
// (all lanes read the same j -> broadcast, no bank conflicts). Each thread
// owns 2 output columns; inner loop is 3 VALU ops per (x,j): v_fmac, v_min,
// v_add. dist = sqrt(min), then sigmoid(pred)*dist, fixed-order block reduce.
// ---------------------------------------------------------------------------
__global__ void __launch_bounds__(256)
bl_row_minplus_loss(const float* __restrict__ pred,
                    const float* __restrict__ g2,
                    const int* __restrict__ hasfg,
                    float* __restrict__ partials) {
    __shared__ float gs[W_];
    __shared__ float red[256];

    const int row  = blockIdx.x;        // b*H + h
    const int b    = row >> 9;
    const int tid  = threadIdx.x;
    const int base = row * W_;

    // prefetch the pred row we touch after the compute loop (global_prefetch_b8)
    __builtin_prefetch(&pred[base + tid], 0, 0);

    gs[tid]       = g2[base + tid];
    gs[tid + 256] = g2[base + tid + 256];
    __syncthreads();

    float d0 = (float)tid;              // x0 - j, maintained incrementally
    float d1 = (float)(tid + 256);
    float m0 = 3.0e38f, m1 = 3.0e38f;
#pragma unroll 8
    for (int j = 0; j < W_; ++j) {
        const float s = gs[j];
        m0 = fminf(m0, __builtin_fmaf(d0, d0, s));
        m1 = fminf(m1, __builtin_fmaf(d1, d1, s));
        d0 -= 1.0f;
        d1 -= 1.0f;
    }

    const float mask = hasfg[b] ? 1.0f : 0.0f;
    const float v0 = pred[base + tid];
    const float v1 = pred[base + tid + 256];
    const float p0 = 1.0f / (1.0f + expf(-v0));
    const float p1 = 1.0f / (1.0f + expf(-v1));
    red[tid] = mask * (p0 * sqrtf(m0) + p1 * sqrtf(m1));
    __syncthreads();

    for (int s = 128; s > 0; s >>= 1) {
        if (tid < s) red[tid] += red[tid + s];
        __syncthreads();
    }
    if (tid == 0) partials[row] = red[0];
}

// ---------------------------------------------------------------------------
// K3: final reduction of the 2048 row partials with V_WMMA_F32_16X16X4_F32.
// Single full wave32 (EXEC all ones, uniform flow). B = all-ones, so
// D[m,n] += sum_k A[m,k]; each wmma folds 64 f32 partials into the f32
// accumulator. Afterwards column n of D equals the grand total when summed
// over m; lane L holds M=0..7 of column L, lane L^16 holds M=8..15.
// ---------------------------------------------------------------------------
__global__ void bl_final_reduce_wmma(const float* __restrict__ partials,
                                     float* __restrict__ out) {
    const int lane = threadIdx.x;       // 0..31, one wave
    v8f acc = {};
    v2f ones;
    ones.x = 1.0f;
    ones.y = 1.0f;

    // 2048 partials = 32 wmma iterations x 64 values; each (lane, reg) slot
    // maps to a unique A-matrix cell, so every partial is counted exactly once.
    for (int it = 0; it < ROWS_ / 64; ++it) {
        v2f a;
        a.x = partials[it * 64 + lane * 2 + 0];
        a.y = partials[it * 64 + lane * 2 + 1];
        acc = __builtin_amdgcn_wmma_f32_16x16x4_f32(
            /*neg_a=*/false, a, /*neg_b=*/false, ones,
            /*c_mod=*/(short)0, acc, /*reuse_a=*/false, /*reuse_b=*/false);
    }

    float s = 0.0f;
#pragma unroll
    for (int r = 0; r < 8; ++r) s += acc[r];
    // column total = rows 0..7 (this half-wave) + rows 8..15 (partner lane)
    const float other = __shfl(s, lane ^ 16, 32);
    const float total = s + other;
    if (lane == 0) out[0] = total / (float)NELEM_;
}

// ---------------------------------------------------------------------------
// Launch
// ---------------------------------------------------------------------------
extern "C" void kernel_launch(void* const* d_in, const int* in_sizes, int n_in,
                              void* d_out, int out_size, void* d_ws, size_t ws_size,
                              hipStream_t stream) {
    const float* pred   = (const float*)d_in[0];   // [4,1,512,512] f32 logits
    const int*   target = (const int*)d_in[1];     // [4,1,512,512] int {0,1}

    // Workspace layout: g2 [NELEM_] f32 | partials [ROWS_] f32 | hasfg [B_] int
    float* g2       = (float*)d_ws;
    float* partials = g2 + NELEM_;
    int*   hasfg    = (int*)(partials + ROWS_);

    bl_init_flags<<<1, 32, 0, stream>>>(hasfg);
    bl_edt_cols<<<(B_ * W_ + 255) / 256, 256, 0, stream>>>(target, g2, hasfg);
    bl_row_minplus_loss<<<ROWS_, 256, 0, stream>>>(pred, g2, hasfg, partials);
    bl_final_reduce_wmma<<<1, 32, 0, stream>>>(partials, (float*)d_out);
}